// Qfnn_50775103373922
// MI455X (gfx1250) — compile-verified
//
#include <hip/hip_runtime.h>
#include <math.h>

typedef __attribute__((ext_vector_type(2))) float v2f;
typedef __attribute__((ext_vector_type(8))) float v8f;

#define THREADS 256

// ---------------------------------------------------------------------------
// Shared per-sample front half: linear -> layernorm -> fuzzy memberships ->
// 8 rule activations. Recomputed in both passes (cheaper than spilling B*8).
// ---------------------------------------------------------------------------
__device__ __forceinline__ void compute_out(
    int tid, const float* __restrict__ x,
    const float* __restrict__ W_lin, const float* __restrict__ b_lin,
    const float* __restrict__ gn_w, const float* __restrict__ gn_b,
    const float* __restrict__ m, const float* __restrict__ theta,
    float out[8])
{
  const float* xr = x + (size_t)tid * 10;
  __builtin_prefetch(xr, 0, 0);          // global_prefetch_b8
  float xv[10];
#pragma unroll
  for (int i = 0; i < 10; ++i) xv[i] = xr[i];
  float h[3];
#pragma unroll
  for (int j = 0; j < 3; ++j) {
    float acc = b_lin[j];
#pragma unroll
    for (int i = 0; i < 10; ++i) acc = fmaf(W_lin[j*10 + i], xv[i], acc);
    h[j] = acc;
  }
  float mu = (h[0] + h[1] + h[2]) * (1.0f/3.0f);
  float d0 = h[0]-mu, d1 = h[1]-mu, d2 = h[2]-mu;
  float var = (d0*d0 + d1*d1 + d2*d2) * (1.0f/3.0f);
  float inv = rsqrtf(var + 1e-5f);
#pragma unroll
  for (int j = 0; j < 3; ++j) h[j] = (h[j]-mu)*inv*gn_w[j] + gn_b[j];

  float p[2][3];
#pragma unroll
  for (int j = 0; j < 3; ++j) {
#pragma unroll
    for (int mem = 0; mem < 2; ++mem) {
      float ts = fabsf(theta[j*2 + mem]) + 1e-6f;
      float dd = h[j] - m[j*2 + mem];
      float f  = expf(-(dd*dd) / (2.0f*ts*ts));
      // sin(arcsin(s))^2 == s^2 for s in [0, 0.99999]
      float sq = fminf(sqrtf(f + 1e-16f), 0.99999f);
      p[mem][j] = sq*sq;
    }
  }
  // IDX[i][j] = bit (2-j) of i
#pragma unroll
  for (int i = 0; i < 8; ++i)
    out[i] = p[(i>>2)&1][0] * p[(i>>1)&1][1] * p[i&1][2];
}

// ---------------------------------------------------------------------------
// Kernel 1: build 8x8 complex unitary of the defuzz circuit (lane c = basis c)
// qubit j <-> bit (2-j) of flat state index.
// ---------------------------------------------------------------------------
__device__ __forceinline__ void qswap(float& a, float& b) { float t=a; a=b; b=t; }

__global__ void k_unitary(const float* __restrict__ w, float* __restrict__ U)
{
  int c = threadIdx.x;
  if (c >= 8) return;
  float re[8], im[8];
#pragma unroll
  for (int i = 0; i < 8; ++i) { re[i] = (i==c) ? 1.f : 0.f; im[i] = 0.f; }

#pragma unroll
  for (int layer = 0; layer < 2; ++layer) {
    // CNOT(q0->q1): ctrl bit2, tgt bit1
#pragma unroll
    for (int i = 0; i < 8; ++i)
      if (((i>>2)&1) && !((i>>1)&1)) { qswap(re[i], re[i|2]); qswap(im[i], im[i|2]); }
    // CNOT(q1->q2): ctrl bit1, tgt bit0
#pragma unroll
    for (int i = 0; i < 8; ++i)
      if (((i>>1)&1) && !(i&1)) { qswap(re[i], re[i|1]); qswap(im[i], im[i|1]); }
    // CNOT(q2->q0): ctrl bit0, tgt bit2
#pragma unroll
    for (int i = 0; i < 8; ++i)
      if ((i&1) && !((i>>2)&1)) { qswap(re[i], re[i|4]); qswap(im[i], im[i|4]); }

#pragma unroll
    for (int j = 0; j < 3; ++j) {
      const int msk = 1 << (2 - j);
      float t0 = 0.5f * w[layer*9 + 3*j + 0];
      float t1 = 0.5f * w[layer*9 + 3*j + 1];
      float t2 = 0.5f * w[layer*9 + 3*j + 2];
      // RX(theta0)
      {
        float cs = cosf(t0), sn = sinf(t0);
#pragma unroll
        for (int i = 0; i < 8; ++i) if (!(i & msk)) {
          int i1 = i | msk;
          float r0 = re[i], i0 = im[i], r1 = re[i1], i1v = im[i1];
          re[i]  = cs*r0 + sn*i1v;  im[i]  = cs*i0 - sn*r1;
          re[i1] = cs*r1 + sn*i0;   im[i1] = cs*i1v - sn*r0;
        }
      }
      // RZ(theta1): e^{-i t} on bit=0, e^{+i t} on bit=1
      {
        float cs = cosf(t1), sn = sinf(t1);
#pragma unroll
        for (int i = 0; i < 8; ++i) {
          float s = (i & msk) ? sn : -sn;
          float r = re[i], v = im[i];
          re[i] = r*cs - v*s;
          im[i] = r*s + v*cs;
        }
      }
      // RX(theta2)
      {
        float cs = cosf(t2), sn = sinf(t2);
#pragma unroll
        for (int i = 0; i < 8; ++i) if (!(i & msk)) {
          int i1 = i | msk;
          float r0 = re[i], i0 = im[i], r1 = re[i1], i1v = im[i1];
          re[i]  = cs*r0 + sn*i1v;  im[i]  = cs*i0 - sn*r1;
          re[i1] = cs*r1 + sn*i0;   im[i1] = cs*i1v - sn*r0;
        }
      }
    }
  }
#pragma unroll
  for (int r = 0; r < 8; ++r) { U[(r*8 + c)*2] = re[r]; U[(r*8 + c)*2 + 1] = im[r]; }
}

// ---------------------------------------------------------------------------
// Kernel 2: pass 1 — per-block partial sums / sums-of-squares of out[8].
// Deterministic (shfl tree + fixed-order LDS combine), no float atomics.
// ---------------------------------------------------------------------------
__global__ void k_pass1(const float* __restrict__ x,
                        const float* __restrict__ W_lin, const float* __restrict__ b_lin,
                        const float* __restrict__ gn_w, const float* __restrict__ gn_b,
                        const float* __restrict__ m, const float* __restrict__ theta,
                        float* __restrict__ partials, int B)
{
  int tid = blockIdx.x * THREADS + threadIdx.x;
  bool valid = tid < B;
  int ctid = valid ? tid : (B - 1);
  float out[8];
  compute_out(ctid, x, W_lin, b_lin, gn_w, gn_b, m, theta, out);

  float vals[16];
#pragma unroll
  for (int i = 0; i < 8; ++i) {
    float o = valid ? out[i] : 0.f;
    vals[i] = o;
    vals[8+i] = o*o;
  }

  __shared__ float wsum[8][16];
  int lane = threadIdx.x & 31;
  int wave = threadIdx.x >> 5;
#pragma unroll
  for (int i = 0; i < 16; ++i) {
    float v = vals[i];
#pragma unroll
    for (int off = 16; off >= 1; off >>= 1) v += __shfl_xor(v, off, 32);
    if (lane == 0) wsum[wave][i] = v;
  }
  __syncthreads();
  if (threadIdx.x < 16) {
    float s = 0.f;
#pragma unroll
    for (int w = 0; w < 8; ++w) s += wsum[w][threadIdx.x];
    partials[(size_t)blockIdx.x * 16 + threadIdx.x] = s;
  }
}

// ---------------------------------------------------------------------------
// Kernel 3: fold partials into fused BN scale/shift (deterministic serial).
// ---------------------------------------------------------------------------
__global__ void k_reduce(const float* __restrict__ partials, int nblocks, int B,
                         const float* __restrict__ bn_gamma, const float* __restrict__ bn_beta,
                         float* __restrict__ scale_shift)
{
  __shared__ float tot[16];
  int q = threadIdx.x;
  if (q < 16) {
    float s = 0.f;
    for (int b = 0; b < nblocks; ++b) s += partials[(size_t)b * 16 + q];
    tot[q] = s;
  }
  __syncthreads();
  if (q < 8) {
    float invB = 1.0f / (float)B;
    float mean = tot[q] * invB;
    float var  = tot[8+q] * invB - mean*mean;
    float sc   = bn_gamma[q] * rsqrtf(var + 1e-5f);
    scale_shift[q]     = sc;
    scale_shift[8 + q] = bn_beta[q] - mean * sc;
  }
}

// ---------------------------------------------------------------------------
// Kernel 4: pass 2 — BN, normalize, U matvec (LDS-broadcast), Z-expectations,
// 3->10 GEMM via V_WMMA_F32_16X16X4_F32. D fragments staged in LDS at the
// final stride-10 layout so the store-back is a pure linear b64 copy.
// ---------------------------------------------------------------------------
__global__ void k_pass2(const float* __restrict__ x,
                        const float* __restrict__ W_lin, const float* __restrict__ b_lin,
                        const float* __restrict__ gn_w, const float* __restrict__ gn_b,
                        const float* __restrict__ m, const float* __restrict__ theta,
                        const float* __restrict__ Uw, const float* __restrict__ scale_shift,
                        const float* __restrict__ W_soft, const float* __restrict__ b_soft,
                        float* __restrict__ y, int B)
{
  __shared__ float Ush[128];
  __shared__ float dzsh[THREADS * 4];
  __shared__ float res[THREADS * 10];   // final row-major 256 x 10 layout

  if (threadIdx.x < 128) Ush[threadIdx.x] = Uw[threadIdx.x];

  int tid = blockIdx.x * THREADS + threadIdx.x;
  bool valid = tid < B;
  int ctid = valid ? tid : (B - 1);
  float out[8];
  compute_out(ctid, x, W_lin, b_lin, gn_w, gn_b, m, theta, out);

  float outn[8], nrm = 0.f;
#pragma unroll
  for (int i = 0; i < 8; ++i) {
    float v = out[i] * scale_shift[i] + scale_shift[8 + i];
    outn[i] = v;
    nrm = fmaf(v, v, nrm);
  }
  float invn = rsqrtf(nrm);
#pragma unroll
  for (int i = 0; i < 8; ++i) outn[i] *= invn;

  __syncthreads();  // Ush ready

  float probs[8];
#pragma unroll
  for (int r = 0; r < 8; ++r) {
    float ar = 0.f, ai = 0.f;
#pragma unroll
    for (int c = 0; c < 8; ++c) {
      ar = fmaf(Ush[(r*8 + c)*2],     outn[c], ar);
      ai = fmaf(Ush[(r*8 + c)*2 + 1], outn[c], ai);
    }
    probs[r] = ar*ar + ai*ai;
  }
  float z0 = (probs[0]+probs[1]+probs[2]+probs[3]) - (probs[4]+probs[5]+probs[6]+probs[7]);
  float z1 = (probs[0]+probs[1]+probs[4]+probs[5]) - (probs[2]+probs[3]+probs[6]+probs[7]);
  float z2 = (probs[0]+probs[2]+probs[4]+probs[6]) - (probs[1]+probs[3]+probs[5]+probs[7]);

  dzsh[threadIdx.x*4 + 0] = z0;
  dzsh[threadIdx.x*4 + 1] = z1;
  dzsh[threadIdx.x*4 + 2] = z2;
  dzsh[threadIdx.x*4 + 3] = 0.f;
  __syncthreads();

  // --- WMMA: D = A(16x4 dz) * B(4x16 W_soft^T) + C(bias) ---
  const int lane = threadIdx.x & 31;
  const int wave = threadIdx.x >> 5;
  const int half = lane >> 4;       // 0: lanes 0-15, 1: lanes 16-31
  const int ln   = lane & 15;       // M index (A) / N index (B, C, D)

  // B fragment: B[k][n] = (n<10 && k<3) ? W_soft[n*3+k] : 0
  // VGPR0 holds K = 2*half, VGPR1 holds K = 2*half + 1
  v2f bfrag;
  bfrag.x = (ln < 10) ? W_soft[ln*3 + 2*half] : 0.f;                 // k = 0 or 2
  bfrag.y = (ln < 10 && half == 0) ? W_soft[ln*3 + 1] : 0.f;         // k = 1 (k=3 -> 0)

  // C fragment: C[m][n] = b_soft[n] for every row
  float bias = (ln < 10) ? b_soft[ln] : 0.f;
  v8f cfrag;
#pragma unroll
  for (int r = 0; r < 8; ++r) cfrag[r] = bias;

  v8f d0frag, d1frag;
  {
    int srow = wave*32 + ln;                      // tile 0 (samples 0-15 of wave)
    v2f afrag;
    afrag.x = dzsh[srow*4 + 2*half];
    afrag.y = dzsh[srow*4 + 2*half + 1];
    d0frag = __builtin_amdgcn_wmma_f32_16x16x4_f32(
        false, afrag, false, bfrag, (short)0, cfrag, false, false);
  }
  {
    int srow = wave*32 + 16 + ln;                 // tile 1 (samples 16-31 of wave)
    v2f afrag;
    afrag.x = dzsh[srow*4 + 2*half];
    afrag.y = dzsh[srow*4 + 2*half + 1];
    d1frag = __builtin_amdgcn_wmma_f32_16x16x4_f32(
        false, afrag, false, bfrag, (short)0, cfrag, false, false);
  }

  // Stage D into LDS in the FINAL stride-10 layout (only valid columns).
  // Lanes 0-15 write 10 consecutive addrs of one row; lanes 16-31 land on
  // banks 16-25 of a row 80 floats away -> conflict-free.
  if (ln < 10) {
#pragma unroll
    for (int r = 0; r < 8; ++r) {
      int row0 = wave*32 +      r + half*8;
      int row1 = wave*32 + 16 + r + half*8;
      res[row0 * 10 + ln] = d0frag[r];
      res[row1 * 10 + ln] = d1frag[r];
    }
  }
  __syncthreads();

  // --- Coalesced output: pure linear copy, zero index math ---
  size_t blockBase = (size_t)blockIdx.x * THREADS * 10;
  bool fullBlock = (size_t)(blockIdx.x + 1) * THREADS <= (size_t)B;
  if (fullBlock) {
    // 2560 floats = 1280 float2 = 5 iterations x 256 threads (b64 ld/st)
    float2* y2 = reinterpret_cast<float2*>(y + blockBase);
    const float2* r2 = reinterpret_cast<const float2*>(res);
#pragma unroll
    for (int it = 0; it < 5; ++it) {
      int i2 = it * THREADS + threadIdx.x;
      y2[i2] = r2[i2];
    }
  } else {
    size_t total = (size_t)B * 10;
    for (int it = 0; it < 10; ++it) {
      int e = it * THREADS + threadIdx.x;
      if (blockBase + e < total)
        y[blockBase + e] = res[e];
    }
  }
}

// ---------------------------------------------------------------------------
extern "C" void kernel_launch(void* const* d_in, const int* in_sizes, int n_in,
                              void* d_out, int out_size, void* d_ws, size_t ws_size,
                              hipStream_t stream)
{
  const float* x        = (const float*)d_in[0];
  const float* W_lin    = (const float*)d_in[1];
  const float* b_lin    = (const float*)d_in[2];
  const float* gn_w     = (const float*)d_in[3];
  const float* gn_b     = (const float*)d_in[4];
  const float* m        = (const float*)d_in[5];
  const float* theta    = (const float*)d_in[6];
  const float* bn_gamma = (const float*)d_in[7];
  const float* bn_beta  = (const float*)d_in[8];
  const float* defuzz_w = (const float*)d_in[9];
  const float* W_soft   = (const float*)d_in[10];
  const float* b_soft   = (const float*)d_in[11];
  float* y = (float*)d_out;

  int B = in_sizes[0] / 10;
  int nblocks = (B + THREADS - 1) / THREADS;

  float* ws          = (float*)d_ws;
  float* U           = ws;            // 128 floats
  float* scale_shift = ws + 128;      // 16 floats
  float* partials    = ws + 144;      // nblocks * 16 floats

  k_unitary<<<1, 32, 0, stream>>>(defuzz_w, U);
  k_pass1<<<nblocks, THREADS, 0, stream>>>(x, W_lin, b_lin, gn_w, gn_b, m, theta, partials, B);
  k_reduce<<<1, 32, 0, stream>>>(partials, nblocks, B, bn_gamma, bn_beta, scale_shift);
  k_pass2<<<nblocks, THREADS, 0, stream>>>(x, W_lin, b_lin, gn_w, gn_b, m, theta,
                                           U, scale_shift, W_soft, b_soft, y, B);
}